// RelationQueryScorer_15805479649711
// MI455X (gfx1250) — compile-verified
//
#include <hip/hip_runtime.h>
#include <math.h>

// B=8, N=128, D=256, H=128, K=8
// Afull: (B*N=1024) x (2H=256) f32 in ws; cols [0,128)=Ai, [128,256)=Aj
// Wqp  : 128 x 16 f32 (cols 8..15 zero-padded) in ws
// bq   : 8 f32 in ws

typedef __attribute__((ext_vector_type(2))) float v2f;
typedef __attribute__((ext_vector_type(8))) float v8f;

// ---------------- Kernel 0: Wq (padded) + bq ----------------
__global__ void wq_kernel(const float* __restrict__ W2, const float* __restrict__ b2,
                          const float* __restrict__ q, float* __restrict__ Wqp,
                          float* __restrict__ bqv) {
  const int h = threadIdx.x; // 0..127
  float s[8];
#pragma unroll
  for (int k = 0; k < 8; ++k) s[k] = 0.0f;
  for (int d = 0; d < 256; ++d) {
    const float w = W2[d * 128 + h]; // W2[d,h]
#pragma unroll
    for (int k = 0; k < 8; ++k) s[k] += w * q[k * 256 + d];
  }
#pragma unroll
  for (int k = 0; k < 8; ++k) Wqp[h * 16 + k] = s[k];
#pragma unroll
  for (int k = 8; k < 16; ++k) Wqp[h * 16 + k] = 0.0f;
  if (h < 8) {
    float t = 0.0f;
    for (int d = 0; d < 256; ++d) t += b2[d] * q[h * 256 + d];
    bqv[h] = t;
  }
}

// ---------------- Kernel 1: Afull = E * [W1a^T | W1b^T] ----------------
// One wave per 16x16 output tile. WMMA f32 16x16x4, K-loop = 64.
// A-frag 16x4 f32: lane(l<16) holds E[row0+l, d+0..1], lane(l+16) holds d+2..3.
// B-frag 4x16 f32: lane(l<16) holds Wc[col0+l, d+0..1], lane(l+16) d+2..3.
__global__ void gemm_A_kernel(const float* __restrict__ E, const float* __restrict__ W1,
                              float* __restrict__ Afull) {
  const int wave = (blockIdx.x * blockDim.x + threadIdx.x) >> 5; // 0..1023
  const int lane = threadIdx.x & 31;
  const int l15 = lane & 15;
  const int hi  = lane >> 4;
  const int tm  = wave >> 4;  // 0..63 row tile
  const int tc  = wave & 15;  // 0..15 col tile
  const int row0 = tm * 16;
  const int col  = tc * 16 + l15;
  // column 'col' of the combined weight matrix: Wc[c,d] = (c<128)? W1[c,d] : W1[c-128, 256+d]
  const float* wcol = (col < 128) ? (W1 + col * 512) : (W1 + (col - 128) * 512 + 256);
  const float* erow = E + (row0 + l15) * 256;
  v8f acc = {};
#pragma unroll 4
  for (int kk = 0; kk < 64; ++kk) {
    const int d = kk * 4 + hi * 2;
    v2f a = *(const v2f*)(erow + d);  // 8B aligned (d even)
    v2f b = *(const v2f*)(wcol + d);
    acc = __builtin_amdgcn_wmma_f32_16x16x4_f32(false, a, false, b, (short)0, acc,
                                                false, false);
  }
  // C layout: vgpr r, lanes0-15 -> row r, lanes16-31 -> row r+8
#pragma unroll
  for (int r = 0; r < 8; ++r) {
    Afull[(row0 + r + hi * 8) * 256 + tc * 16 + l15] = acc[r];
  }
}

// ---------------- Kernel 2: fused relu-MLP + query score + sigmoid + mask ----------------
// One wave per (b, i, j-tile of 16). C[k, j] = WqT(16x128, rows 8..15 zero) x hidden^T(128x16)
// hidden^T fragment built in registers: relu(Aj[b,j,h] + Ai[b,i,h] + b1[h]).
__global__ void score_kernel(const float* __restrict__ Afull, const float* __restrict__ b1,
                             const float* __restrict__ Wqp, const float* __restrict__ bqv,
                             float* __restrict__ out) {
  __shared__ float sWq[128 * 16]; // 8 KB
  // stage padded Wq into LDS (256 threads x 8 floats)
  {
    const int t = threadIdx.x;
#pragma unroll
    for (int u = 0; u < 8; ++u) sWq[t + u * 256] = Wqp[t + u * 256];
  }
  __syncthreads();

  const int wave = (blockIdx.x * blockDim.x + threadIdx.x) >> 5; // 0..8191
  const int lane = threadIdx.x & 31;
  const int l15 = lane & 15;
  const int hi  = lane >> 4;
  const int jt  = wave & 7;          // j tile
  const int i   = (wave >> 3) & 127; // row i
  const int b   = wave >> 10;        // batch
  const int j   = jt * 16 + l15;     // this lane's j (B-frag column)

  const float* ai = Afull + (b * 128 + i) * 256;       // Ai row (Ai part, cols 0..127)
  const float* aj = Afull + (b * 128 + j) * 256 + 128; // Aj row (per-lane j)

  v8f acc = {};
#pragma unroll 4
  for (int kk = 0; kk < 32; ++kk) {
    const int h = kk * 4 + hi * 2;
    // A-frag: WqT[k=l15, h..h+1] ; zero-padded rows k>=8 handled by padding
    v2f a;
    a.x = sWq[h * 16 + l15];
    a.y = sWq[(h + 1) * 16 + l15];
    // B-frag: hidden^T[h..h+1, j] computed on the fly
    v2f xa = *(const v2f*)(aj + h);
    v2f xi = *(const v2f*)(ai + h);
    v2f bb = *(const v2f*)(b1 + h);
    v2f hv;
    hv.x = fmaxf(xa.x + xi.x + bb.x, 0.0f);
    hv.y = fmaxf(xa.y + xi.y + bb.y, 0.0f);
    acc = __builtin_amdgcn_wmma_f32_16x16x4_f32(false, a, false, hv, (short)0, acc,
                                                false, false);
  }

  // C layout: vgpr r, lanes0-15 hold C[k=r, j=jt*16+lane]; rows 8..15 (lanes16-31) are zero.
  if (hi == 0) {
#pragma unroll
    for (int r = 0; r < 8; ++r) {
      float v = acc[r] + bqv[r];
      float s = 1.0f / (1.0f + __expf(-v));
      if (i == j) s = 0.0f; // diagonal mask
      out[(((b * 8 + r) * 128) + i) * 128 + j] = s; // contiguous over j per 16 lanes
    }
  }
}

extern "C" void kernel_launch(void* const* d_in, const int* in_sizes, int n_in,
                              void* d_out, int out_size, void* d_ws, size_t ws_size,
                              hipStream_t stream) {
  const float* E  = (const float*)d_in[0]; // (8,128,256)
  const float* W1 = (const float*)d_in[1]; // (128,512)
  const float* b1 = (const float*)d_in[2]; // (128)
  const float* W2 = (const float*)d_in[3]; // (256,128)
  const float* b2 = (const float*)d_in[4]; // (256)
  const float* q  = (const float*)d_in[5]; // (8,256)
  float* out = (float*)d_out;              // (8,8,128,128)

  float* Afull = (float*)d_ws;             // 1024*256 = 262144 floats
  float* Wqp   = Afull + 262144;           // 128*16   = 2048 floats
  float* bqv   = Wqp + 2048;               // 8 floats

  wq_kernel<<<dim3(1), dim3(128), 0, stream>>>(W2, b2, q, Wqp, bqv);
  gemm_A_kernel<<<dim3(128), dim3(256), 0, stream>>>(E, W1, Afull);
  score_kernel<<<dim3(1024), dim3(256), 0, stream>>>(Afull, b1, Wqp, bqv, out);
}